// Unpool_4320737100488
// MI455X (gfx1250) — compile-verified
//
#include <hip/hip_runtime.h>

// Unpool / row-scatter:  out[N,D] = 0;  out[idx[k], :] = h[k, :]
// N=10000, K=5000, D=512, fp32.  Zero FLOPs -> pure HBM-bandwidth problem.
// Strategy: build inverse map inv[N] in d_ws (inv[idx[k]] = k), then write
// every output row exactly once with b128 stores (gather from h or zeros).
// Min traffic ~31 MB -> ~1.33 us at 23.3 TB/s.

typedef float float4_t __attribute__((ext_vector_type(4)));

// ---- pass 1a: inv[i] = -1 ------------------------------------------------
__global__ void unpool_fill_inv(int* __restrict__ inv, int n) {
    int i = blockIdx.x * blockDim.x + threadIdx.x;
    if (i < n) inv[i] = -1;
}

// ---- pass 1b: inv[idx[k]] = k  (indices unique per reference) ------------
__global__ void unpool_build_inv(const int* __restrict__ idx,
                                 int* __restrict__ inv, int k, int n) {
    int i = blockIdx.x * blockDim.x + threadIdx.x;
    if (i < k) {
        int t = idx[i];
        if (t >= 0 && t < n) inv[t] = i;
    }
}

// ---- pass 2: one block per output row; write each row exactly once -------
// block = 128 threads (4 wave32), each thread covers col, col+128, ... in
// float4 units.  global_load_b128 / global_store_b128 throughout.
__global__ void unpool_gather_rows(const float4_t* __restrict__ h,
                                   const int* __restrict__ inv,
                                   float4_t* __restrict__ out, int d4) {
    const int row = blockIdx.x;
    const int src = inv[row];                 // broadcast within block (L0 hit)
    float4_t* __restrict__ orow = out + (long long)row * d4;
    if (src >= 0) {
        const float4_t* __restrict__ hrow = h + (long long)src * d4;
        for (int c = threadIdx.x; c < d4; c += blockDim.x)
            orow[c] = hrow[c];
    } else {
        const float4_t z = {0.f, 0.f, 0.f, 0.f};
        for (int c = threadIdx.x; c < d4; c += blockDim.x)
            orow[c] = z;
    }
}

// ---- fallback (ws too small): memset + direct scatter --------------------
__global__ void unpool_zero_out(float4_t* __restrict__ out, long long total4) {
    long long i = (long long)blockIdx.x * blockDim.x + threadIdx.x;
    const float4_t z = {0.f, 0.f, 0.f, 0.f};
    if (i < total4) out[i] = z;
}

__global__ void unpool_scatter_rows(const float4_t* __restrict__ h,
                                    const int* __restrict__ idx,
                                    float4_t* __restrict__ out, int d4, int n) {
    const int k = blockIdx.x;
    const int t = idx[k];
    if (t < 0 || t >= n) return;
    const float4_t* __restrict__ hrow = h + (long long)k * d4;
    float4_t* __restrict__ orow = out + (long long)t * d4;
    for (int c = threadIdx.x; c < d4; c += blockDim.x)
        orow[c] = hrow[c];
}

extern "C" void kernel_launch(void* const* d_in, const int* in_sizes, int n_in,
                              void* d_out, int out_size, void* d_ws, size_t ws_size,
                              hipStream_t stream) {
    // inputs: [0] g (N*N f32, shape-only), [1] h (K*D f32), [2] idx (K int)
    const float* h   = (const float*)d_in[1];
    const int*   idx = (const int*)d_in[2];

    const int K  = in_sizes[2];
    const int D  = in_sizes[1] / K;      // 512
    const int N  = out_size / D;         // 10000
    const int d4 = D / 4;                // 128 float4 per row (D % 4 == 0 here)

    float* out = (float*)d_out;

    if (ws_size >= (size_t)N * sizeof(int)) {
        int* inv = (int*)d_ws;
        // pass 1: inverse index map (tiny: ~60 KB of traffic total)
        unpool_fill_inv <<<(N + 255) / 256, 256, 0, stream>>>(inv, N);
        unpool_build_inv<<<(K + 255) / 256, 256, 0, stream>>>(idx, inv, K, N);
        // pass 2: single sweep over output, each byte written exactly once
        unpool_gather_rows<<<N, 128, 0, stream>>>(
            (const float4_t*)h, inv, (float4_t*)out, d4);
    } else {
        // fallback: zero then scatter (extra 20 MB of zero-stores)
        const long long total4 = (long long)N * d4;
        unpool_zero_out<<<(unsigned)((total4 + 255) / 256), 256, 0, stream>>>(
            (float4_t*)out, total4);
        unpool_scatter_rows<<<K, 128, 0, stream>>>(
            (const float4_t*)h, idx, (float4_t*)out, d4, N);
    }
}